// QuantizedEmbedding_1271310320438
// MI455X (gfx1250) — compile-verified
//
#include <hip/hip_runtime.h>
#include <hip/hip_fp16.h>
#include <hip/hip_bf16.h>

#define THREADS 256
#define DDIM    4096
#define HALF_D  2048      // packed int32 elements per row
#define NGROUPS 128

typedef int v4i __attribute__((ext_vector_type(4)));   // native vector -> OK for nontemporal builtin

__global__ __launch_bounds__(THREADS)
void qembed_dequant_kernel(const int* __restrict__ input,
                           const int* __restrict__ wpacked,
                           const __half* __restrict__ wscale,
                           const __half* __restrict__ wmin,
                           __hip_bfloat16* __restrict__ out)
{
    // Full 8KB packed row staged here by the async global->LDS DMA engine.
    __shared__ v4i tile[2 * THREADS];

    const int row = blockIdx.x;
    const int t   = threadIdx.x;

    const long long ridx = (long long)input[row];   // uniform -> scalar load

    // ---- CDNA5 async global->LDS DMA: stage this thread's 32B of the packed row.
    // GVS addressing: SGPR 64-bit row base + per-lane 32-bit byte offset.
    // offset:16 applies to both the global and LDS addresses, so one operand
    // set covers both 16B halves of the 32B chunk.
    {
        unsigned long long gbase = (unsigned long long)(wpacked + ridx * HALF_D);
        unsigned int voff = (unsigned int)(t * 32);
        unsigned int ldsa = (unsigned int)(unsigned long long)(&tile[2 * t]);
        asm volatile(
            "global_load_async_to_lds_b128 %0, %1, %2\n\t"
            "global_load_async_to_lds_b128 %0, %1, %2 offset:16"
            :
            : "v"(ldsa), "v"(voff), "s"(gbase)
            : "memory");
    }

    // ---- Overlap the DMA with the fp16 scale/min fetch for this thread's group.
    const int g = t >> 1;
    const __half  s  = wscale[ridx * NGROUPS + g];
    const __half  m  = wmin [ridx * NGROUPS + g];
    const __half2 s2 = __half2half2(s);
    const __half2 m2 = __half2half2(m);

    // ---- Wait for this wave's async copies to land in LDS (per-wave ASYNCcnt,
    // each thread only reads bytes its own wave wrote -> no barrier needed).
    asm volatile("s_wait_asynccnt 0" ::: "memory");

    const v4i pa = tile[2 * t + 0];
    const v4i pb = tile[2 * t + 1];
    const int pk[8] = {pa.x, pa.y, pa.z, pa.w, pb.x, pb.y, pb.z, pb.w};

    // Dequant matches reference rounding: fp16 mul, fp16 sub, then RNE to bf16.
    unsigned int w[8];
#pragma unroll
    for (int i = 0; i < 8; ++i) {
        const int p = pk[i];
        const __half2 v = __halves2half2(
            __ushort2half_rn((unsigned short)(p & 0xF)),          // low nibble -> even elem
            __ushort2half_rn((unsigned short)((p >> 4) & 0xF)));  // high nibble -> odd elem
        const __half2 r = __hsub2(__hmul2(v, s2), m2);
        const __hip_bfloat16 b0 = __float2bfloat16(__low2float(r));
        const __hip_bfloat16 b1 = __float2bfloat16(__high2float(r));
        w[i] = (unsigned int)__builtin_bit_cast(unsigned short, b0)
             | ((unsigned int)__builtin_bit_cast(unsigned short, b1) << 16);
    }

    // ---- Non-temporal B128 stores: 256MB write-once output, keep it out of L2
    // so gathered weight rows retain cache residency.
    __hip_bfloat16* orow = out + (size_t)row * DDIM + (size_t)t * 16;
    const v4i o0 = {(int)w[0], (int)w[1], (int)w[2], (int)w[3]};
    const v4i o1 = {(int)w[4], (int)w[5], (int)w[6], (int)w[7]};
    __builtin_nontemporal_store(o0, (v4i*)orow);
    __builtin_nontemporal_store(o1, ((v4i*)orow) + 1);
}

extern "C" void kernel_launch(void* const* d_in, const int* in_sizes, int n_in,
                              void* d_out, int out_size, void* d_ws, size_t ws_size,
                              hipStream_t stream) {
    const int*    input   = (const int*)d_in[0];     // [B*S] token ids
    const int*    wpacked = (const int*)d_in[1];     // [V, D/2] packed nibbles in int32
    const __half* wscale  = (const __half*)d_in[2];  // [V, NGROUPS] fp16
    const __half* wmin    = (const __half*)d_in[3];  // [V, NGROUPS] fp16
    __hip_bfloat16* out   = (__hip_bfloat16*)d_out;  // [B*S, D] bf16

    const int n = in_sizes[0];                       // 32768 rows
    qembed_dequant_kernel<<<dim3((unsigned)n), dim3(THREADS), 0, stream>>>(
        input, wpacked, wscale, wmin, out);
}